// SnConv3to2_5059471475001
// MI455X (gfx1250) — compile-verified
//
#include <hip/hip_runtime.h>

// SnConv3to2 for MI455X (gfx1250, wave32).
// B=8, n=64, C=32, O=32.  x: [B,n,n,n,C] f32 (256 MB) -> out: [B,n,n,O] f32.
// Memory-bound: single streaming pass over x produces all contractions
// deterministically (no atomics); final mix GEMM uses V_WMMA_F32_16X16X4_F32.

typedef __attribute__((ext_vector_type(2))) float v2f;
typedef __attribute__((ext_vector_type(8))) float v8f;

// ---- workspace layout (float offsets) ----
static constexpr size_t S1P  = 0;            // [B,8,n,n,C] partial sums over i-groups
static constexpr size_t S2P  = 8388608;      // [B,8,n,n,C] partial sums over j-groups
static constexpr size_t S3M  = 16777216;     // [B,n,n,C] mean over k
static constexpr size_t D12  = 17825792;     // [B,n,n,C] x[b,i,i,k,c]
static constexpr size_t D13  = 18874368;     // [B,n,n,C] x[b,i,j,i,c]
static constexpr size_t D23  = 19922944;     // [B,n,n,C] x[b,a,i,i,c]
static constexpr size_t TRI  = 20971520;     // [B,n,C]   x[b,i,i,i,c]
static constexpr size_t S1M  = 20987904;     // [B,n,n,C] mean over i
static constexpr size_t S2M  = 22036480;     // [B,n,n,C] mean over j
static constexpr size_t A1O  = 23085056;     // [B,n,10C]
static constexpr size_t A0O  = 23248896;     // [B,5C]
static constexpr size_t R12O = 23250176;     // [B,n,O]  a1@W12+b12
static constexpr size_t R12TO= 23266560;     // [B,n,O]  a1@W12t+b12t
static constexpr size_t FORO = 23282944;     // [B,n,O]  a1@W11+b11 + a0@W01+b01
static constexpr size_t R02O = 23299328;     // [B,O]    a0@W02+b02+b22
static constexpr size_t WS_FLOATS = 23299584; // 93,198,336 bytes needed

// ===================== Pass 1: stream x once ============================
// grid (jt, it, b) = (8,8,8); 256 threads. Each block reads the 8x8 (i,j)
// tile of rows x[b,i,j,:,:] (64 rows x 8KB, coalesced).
// Thread t owns row elements e = t + 256*m (m=0..7); c = t&31, k = t/32 + 8m.
__global__ void k1_contract(const float* __restrict__ x,
                            float* __restrict__ s1p, float* __restrict__ s2p,
                            float* __restrict__ s3m, float* __restrict__ d12,
                            float* __restrict__ d13, float* __restrict__ d23,
                            float* __restrict__ tri) {
  __shared__ float lds[7 * 2048 + 256];   // 58368 B: s2 acc (ii<7) + s3 scratch
  const int t  = threadIdx.x;
  const int jt = blockIdx.x, it = blockIdx.y, b = blockIdx.z;
  const int kg = t >> 5, c = t & 31;

  for (int idx = t; idx < 7 * 2048; idx += 256) lds[idx] = 0.f;
  float s2r[8];
  #pragma unroll
  for (int m = 0; m < 8; ++m) s2r[m] = 0.f;
  __syncthreads();

  for (int jj = 0; jj < 8; ++jj) {
    const int j = jt * 8 + jj;
    float s1a[8];
    #pragma unroll
    for (int m = 0; m < 8; ++m) s1a[m] = 0.f;

    for (int ii = 0; ii < 8; ++ii) {
      const int i = it * 8 + ii;
      const float* row = x + (((size_t)((b * 64 + i) * 64 + j)) << 11);
      float v[8];
      #pragma unroll
      for (int m = 0; m < 8; ++m) v[m] = row[t + (m << 8)];

      float p = 0.f;
      #pragma unroll
      for (int m = 0; m < 8; ++m) { s1a[m] += v[m]; p += v[m]; }

      if (ii < 7) {
        #pragma unroll
        for (int m = 0; m < 8; ++m) lds[ii * 2048 + t + (m << 8)] += v[m];
      } else {
        #pragma unroll
        for (int m = 0; m < 8; ++m) s2r[m] += v[m];
      }

      lds[14336 + t] = p;                 // s3 scratch
      __syncthreads();
      if (t < 32) {
        float tot = 0.f;
        #pragma unroll
        for (int g = 0; g < 8; ++g) tot += lds[14336 + g * 32 + t];
        s3m[(((size_t)((b * 64 + i) * 64 + j)) << 5) + t] = tot * (1.f / 64.f);
      }
      // diagonals (gathers, written exactly once across the grid)
      if ((i & 7) == kg) d13[(((size_t)((b * 64 + i) * 64 + j)) << 5) + c] = v[i >> 3];
      if ((j & 7) == kg) d23[(((size_t)((b * 64 + i) * 64 + j)) << 5) + c] = v[j >> 3];
      if (i == j) {
        #pragma unroll
        for (int m = 0; m < 8; ++m)
          d12[(((size_t)(b * 64 + i)) << 11) + t + (m << 8)] = v[m];
        if ((i & 7) == kg) tri[((b << 6) + i) * 32 + c] = v[i >> 3];
      }
      __syncthreads();                    // protect s3 scratch for next row
    }
    float* dst = s1p + (((size_t)((b * 8 + it) * 64 + j)) << 11);
    #pragma unroll
    for (int m = 0; m < 8; ++m) dst[t + (m << 8)] = s1a[m];
  }
  // write s2 partials (each thread reads only its own LDS slots)
  for (int ii = 0; ii < 8; ++ii) {
    float* dst = s2p + (((size_t)((b * 8 + jt) * 64 + it * 8 + ii)) << 11);
    if (ii < 7) {
      #pragma unroll
      for (int m = 0; m < 8; ++m) dst[t + (m << 8)] = lds[ii * 2048 + t + (m << 8)];
    } else {
      #pragma unroll
      for (int m = 0; m < 8; ++m) dst[t + (m << 8)] = s2r[m];
    }
  }
}

// ===================== Pass 2a: reduce partials -> means =================
__global__ void k2a_reduce(const float* __restrict__ s1p, const float* __restrict__ s2p,
                           float* __restrict__ s1m, float* __restrict__ s2m) {
  const int t = threadIdx.x;
  const int u = blockIdx.x & 63, b = blockIdx.x >> 6;
  #pragma unroll
  for (int m = 0; m < 8; ++m) {
    const int e = t + (m << 8);
    float a = 0.f, s = 0.f;
    #pragma unroll
    for (int g = 0; g < 8; ++g) {
      a += s1p[(((size_t)((b * 8 + g) * 64 + u)) << 11) + e];
      s += s2p[(((size_t)((b * 8 + g) * 64 + u)) << 11) + e];
    }
    s1m[(((size_t)(b * 64 + u)) << 11) + e] = a * (1.f / 64.f);
    s2m[(((size_t)(b * 64 + u)) << 11) + e] = s * (1.f / 64.f);
  }
}

// ===================== Pass 2b: a1 [B,n,10C] =============================
__global__ void k2b_a1(const float* __restrict__ s1m, const float* __restrict__ s3m,
                       const float* __restrict__ d12, const float* __restrict__ d13,
                       const float* __restrict__ d23, const float* __restrict__ tri,
                       float* __restrict__ a1) {
  __shared__ float scr[256];
  const int t = threadIdx.x, c = t & 31, g = t >> 5;
  const int b = blockIdx.x >> 6, u = blockIdx.x & 63;
  float* dst = a1 + (size_t)(b * 64 + u) * 320;

  auto red = [&](float partial, int slot) {
    scr[t] = partial;
    __syncthreads();
    if (t < 32) {
      float tot = 0.f;
      #pragma unroll
      for (int gg = 0; gg < 8; ++gg) tot += scr[gg * 32 + t];
      dst[slot * 32 + t] = tot * (1.f / 64.f);
    }
    __syncthreads();
  };
  float acc;
  // p0: mean(x,(1,2))[b,u,c] = (1/64) sum_j s1m[b,j,u,c]
  acc = 0.f; for (int s = 0; s < 8; ++s) { int v = g + 8 * s;
    acc += s1m[(((size_t)((b * 64 + v) * 64 + u)) << 5) + c]; } red(acc, 0);
  // p1: mean(x,(1,3))[b,u,c] = (1/64) sum_k s1m[b,u,k,c]
  acc = 0.f; for (int s = 0; s < 8; ++s) { int v = g + 8 * s;
    acc += s1m[(((size_t)((b * 64 + u) * 64 + v)) << 5) + c]; } red(acc, 1);
  // p2: mean(x,(2,3))[b,u,c] = (1/64) sum_j s3m[b,u,j,c]
  acc = 0.f; for (int s = 0; s < 8; ++s) { int v = g + 8 * s;
    acc += s3m[(((size_t)((b * 64 + u) * 64 + v)) << 5) + c]; } red(acc, 2);
  // p3/p4: d12 means over axis1 / axis2
  acc = 0.f; for (int s = 0; s < 8; ++s) { int v = g + 8 * s;
    acc += d12[(((size_t)(b * 64 + v)) << 11) + u * 32 + c]; } red(acc, 3);
  acc = 0.f; for (int s = 0; s < 8; ++s) { int v = g + 8 * s;
    acc += d12[(((size_t)(b * 64 + u)) << 11) + v * 32 + c]; } red(acc, 4);
  // p5/p6: d13
  acc = 0.f; for (int s = 0; s < 8; ++s) { int v = g + 8 * s;
    acc += d13[(((size_t)((b * 64 + v) * 64 + u)) << 5) + c]; } red(acc, 5);
  acc = 0.f; for (int s = 0; s < 8; ++s) { int v = g + 8 * s;
    acc += d13[(((size_t)((b * 64 + u) * 64 + v)) << 5) + c]; } red(acc, 6);
  // p7/p8: d23
  acc = 0.f; for (int s = 0; s < 8; ++s) { int v = g + 8 * s;
    acc += d23[(((size_t)((b * 64 + v) * 64 + u)) << 5) + c]; } red(acc, 7);
  acc = 0.f; for (int s = 0; s < 8; ++s) { int v = g + 8 * s;
    acc += d23[(((size_t)((b * 64 + u) * 64 + v)) << 5) + c]; } red(acc, 8);
  // p9: triple diagonal (copy)
  if (t < 32) dst[9 * 32 + t] = tri[(b * 64 + u) * 32 + t];
}

// ===================== Pass 2c: a0 [B,5C] ================================
__global__ void k2c_a0(const float* __restrict__ a1, float* __restrict__ a0) {
  const int b = blockIdx.x, t = threadIdx.x;
  if (t < 160) {
    const int s = t >> 5, c = t & 31;
    const int slot = (s == 0) ? 0 : (2 * s + 1);   // {0,3,5,7,9}
    float acc = 0.f;
    for (int u = 0; u < 64; ++u)
      acc += a1[(size_t)(b * 64 + u) * 320 + slot * 32 + c];
    a0[b * 160 + s * 32 + c] = acc * (1.f / 64.f);
  }
}

// ===================== Pass 3a: row mixes ================================
__global__ void k3a_rows(const float* __restrict__ a1, const float* __restrict__ a0,
                         const float* __restrict__ W12, const float* __restrict__ b12,
                         const float* __restrict__ W12t, const float* __restrict__ b12t,
                         const float* __restrict__ W11, const float* __restrict__ b11,
                         const float* __restrict__ W01, const float* __restrict__ b01,
                         const float* __restrict__ W02, const float* __restrict__ b02,
                         const float* __restrict__ b22,
                         float* __restrict__ r12, float* __restrict__ r12t,
                         float* __restrict__ forow, float* __restrict__ r02) {
  const int row = blockIdx.x;            // b*64+p
  const int b = row >> 6, p = row & 63;
  const int n = threadIdx.x;             // 32 threads = O
  const float* a1r = a1 + (size_t)row * 320;
  const float* a0r = a0 + b * 160;
  float v12 = b12[n], v12t = b12t[n], vfo = b11[n] + b01[n];
  for (int k = 0; k < 320; ++k) {
    const float av = a1r[k];
    v12  += av * W12 [k * 32 + n];
    v12t += av * W12t[k * 32 + n];
    vfo  += av * W11 [k * 32 + n];
  }
  for (int k = 0; k < 160; ++k) vfo += a0r[k] * W01[k * 32 + n];
  r12  [row * 32 + n] = v12;
  r12t [row * 32 + n] = v12t;
  forow[row * 32 + n] = vfo;
  if (p == 0) {
    float v02 = b02[n] + b22[n];         // fold b22 in here
    for (int k = 0; k < 160; ++k) v02 += a0r[k] * W02[k * 32 + n];
    r02[b * 32 + n] = v02;
  }
}

// ===================== Pass 3b: WMMA mix GEMM ============================
// out[b,p,q,:] = a2_w_t[b,p,q,:] @ W22 (K=384, N=32) + additive terms.
// Each wave: one 16(M)x32(N) tile via V_WMMA_F32_16X16X4_F32, 96 K-steps.
__global__ void k3b_mix(const float* __restrict__ s1m, const float* __restrict__ s2m,
                        const float* __restrict__ s3m, const float* __restrict__ d12,
                        const float* __restrict__ d13, const float* __restrict__ d23,
                        const float* __restrict__ W22,
                        const float* __restrict__ r12, const float* __restrict__ r12t,
                        const float* __restrict__ r02, const float* __restrict__ forow,
                        float* __restrict__ out) {
  __shared__ float w[384 * 32];          // 48KB: W22 tile
  const int t = threadIdx.x;
  for (int idx = t; idx < 384 * 32; idx += 256) w[idx] = W22[idx];
  __syncthreads();

  const int lane = t & 31;
  const int R = (blockIdx.x * 8 + (t >> 5)) * 16;   // first row of tile
  // A-fragment row for this lane (fixed over the K loop)
  const int MA = lane & 15;
  const int kh = lane >> 4;              // lanes 0-15 hold K0/K1, 16-31 K2/K3
  const int rowA = R + MA;
  const int bA = rowA >> 12, pA = (rowA >> 6) & 63, qA = rowA & 63;
  const size_t offPQ = ((size_t)((bA * 64 + pA) * 64 + qA)) << 5;
  const size_t offQP = ((size_t)((bA * 64 + qA) * 64 + pA)) << 5;
  const float* comp[6] = { s1m, s2m, s3m, d12, d13, d23 };
  const int n0 = lane & 15;

  v8f acc0 = {};                         // N = 0..15
  v8f acc1 = {};                         // N = 16..31
  #pragma unroll
  for (int f = 0; f < 12; ++f) {         // 12 feature blocks of C=32
    const float* A = (f < 6) ? (comp[f] + offPQ) : (comp[f - 6] + offQP);
    #pragma unroll
    for (int kk = 0; kk < 32; kk += 4) {
      v2f a = *(const v2f*)(A + kk + 2 * kh);
      const int kcol = f * 32 + kk + 2 * kh;
      v2f b0 = { w[kcol * 32 + n0],      w[(kcol + 1) * 32 + n0] };
      v2f b1 = { w[kcol * 32 + n0 + 16], w[(kcol + 1) * 32 + n0 + 16] };
      acc0 = __builtin_amdgcn_wmma_f32_16x16x4_f32(false, a, false, b0,
                                                   (short)0, acc0, false, false);
      acc1 = __builtin_amdgcn_wmma_f32_16x16x4_f32(false, a, false, b1,
                                                   (short)0, acc1, false, false);
    }
  }
  // store: VGPR r holds M=r (lanes 0-15) / M=r+8 (lanes 16-31), N=lane&15
  #pragma unroll
  for (int r = 0; r < 8; ++r) {
    const int M = r + ((lane >> 4) << 3);
    const int row = R + M;
    const int b = row >> 12, p = (row >> 6) & 63, q = row & 63;
    float add0 = r12[((b << 6) + p) * 32 + n0] + r12t[((b << 6) + q) * 32 + n0]
               + r02[(b << 5) + n0];
    float add1 = r12[((b << 6) + p) * 32 + n0 + 16] + r12t[((b << 6) + q) * 32 + n0 + 16]
               + r02[(b << 5) + n0 + 16];
    if (p == q) {
      add0 += forow[((b << 6) + p) * 32 + n0];
      add1 += forow[((b << 6) + p) * 32 + n0 + 16];
    }
    out[((size_t)row << 5) + n0]      = acc0[r] + add0;
    out[((size_t)row << 5) + n0 + 16] = acc1[r] + add1;
  }
}

// ===================== launch ===========================================
extern "C" void kernel_launch(void* const* d_in, const int* in_sizes, int n_in,
                              void* d_out, int out_size, void* d_ws, size_t ws_size,
                              hipStream_t stream) {
  (void)in_sizes; (void)n_in; (void)out_size;
  if (ws_size < WS_FLOATS * sizeof(float)) return;   // need ~89 MiB scratch

  const float* x    = (const float*)d_in[0];
  const float* W22  = (const float*)d_in[1];
  const float* b22  = (const float*)d_in[2];
  const float* W12  = (const float*)d_in[3];
  const float* b12  = (const float*)d_in[4];
  const float* W12t = (const float*)d_in[5];
  const float* b12t = (const float*)d_in[6];
  const float* W02  = (const float*)d_in[7];
  const float* b02  = (const float*)d_in[8];
  const float* W11  = (const float*)d_in[9];
  const float* b11  = (const float*)d_in[10];
  const float* W01  = (const float*)d_in[11];
  const float* b01  = (const float*)d_in[12];

  float* ws = (float*)d_ws;
  float* s1p = ws + S1P;   float* s2p = ws + S2P;
  float* s3m = ws + S3M;   float* d12 = ws + D12;
  float* d13 = ws + D13;   float* d23 = ws + D23;
  float* tri = ws + TRI;   float* s1m = ws + S1M;
  float* s2m = ws + S2M;   float* a1  = ws + A1O;
  float* a0  = ws + A0O;   float* r12 = ws + R12O;
  float* r12t= ws + R12TO; float* forow = ws + FORO;
  float* r02 = ws + R02O;

  k1_contract<<<dim3(8, 8, 8), 256, 0, stream>>>(x, s1p, s2p, s3m, d12, d13, d23, tri);
  k2a_reduce <<<512, 256, 0, stream>>>(s1p, s2p, s1m, s2m);
  k2b_a1     <<<512, 256, 0, stream>>>(s1m, s3m, d12, d13, d23, tri, a1);
  k2c_a0     <<<8, 256, 0, stream>>>(a1, a0);
  k3a_rows   <<<512, 32, 0, stream>>>(a1, a0, W12, b12, W12t, b12t,
                                      W11, b11, W01, b01, W02, b02, b22,
                                      r12, r12t, forow, r02);
  k3b_mix    <<<256, 256, 0, stream>>>(s1m, s2m, s3m, d12, d13, d23, W22,
                                       r12, r12t, r02, forow, (float*)d_out);
}